// LinearAttention_1726576857297
// MI455X (gfx1250) — compile-verified
//
#include <hip/hip_runtime.h>
#include <hip/hip_bf16.h>
#include <math.h>

typedef __attribute__((ext_vector_type(2))) float v2f;
typedef __attribute__((ext_vector_type(8))) float v8f;

#define CDIM   512
#define NPIX   4096     // 64*64 per batch
#define NB     16
#define HEADS  8
#define DHEAD  64
#define OQKV   1536
#define SQRT_C 22.62741699796952f   // sqrt(512)
#define EPS    1e-12f

__device__ __forceinline__ v8f wmma_f32(v2f a, v2f b, v8f c) {
    // V_WMMA_F32_16X16X4_F32 : D = A(16x4,f32) * B(4x16,f32) + C(16x16,f32)
    return __builtin_amdgcn_wmma_f32_16x16x4_f32(
        /*neg_a=*/false, a, /*neg_b=*/false, b,
        /*c_mod=*/(short)0, c, /*reuse_a=*/false, /*reuse_b=*/false);
}

// ---------------------------------------------------------------- kernel 0
// Wg[o][c] = w_qkv[o][c] * gamma1[c]   (fold rmsnorm gamma into weights)
__global__ __launch_bounds__(256) void prep_wg(const float* __restrict__ wqkv,
                                               const float* __restrict__ gamma1,
                                               float* __restrict__ wg) {
    int i = blockIdx.x * 256 + threadIdx.x;           // OQKV*CDIM elements
    wg[i] = wqkv[i] * gamma1[i & (CDIM - 1)];
}

// ---------------------------------------------------------------- kernel 1
// per-pixel scale s1[p] = sqrt(512)/max(||x[:,p]||, eps)
__global__ __launch_bounds__(256) void norm1(const float* __restrict__ x,
                                             float* __restrict__ s1) {
    int p = blockIdx.x * 256 + threadIdx.x;           // 65536 pixels
    int b = p >> 12, n = p & (NPIX - 1);
    const float* xb = x + (size_t)b * CDIM * NPIX + n;
    float acc = 0.f;
    for (int c = 0; c < CDIM; ++c) {
        float v = xb[(size_t)c * NPIX];
        acc = fmaf(v, v, acc);
    }
    s1[p] = SQRT_C / fmaxf(sqrtf(acc), EPS);
}

// ---------------------------------------------------------------- kernel 2
// qkv[b][o][n] = s1[b,n] * sum_c Wg[o][c] * x[b][c][n]
// wave tile: 64(O) x 32(N), K=512 in steps of 4 (f32 WMMA).
// Each A fragment is reused by 2 B fragments -> ~1 VMEM op per WMMA.
__global__ __launch_bounds__(256) void gemm_qkv(const float* __restrict__ x,
                                                const float* __restrict__ wg,
                                                const float* __restrict__ s1,
                                                float* __restrict__ qkv) {
    int wid  = blockIdx.x * 8 + (threadIdx.x >> 5);   // 49152 waves
    int lane = threadIdx.x & 31;
    int m = lane & 15, h = lane >> 4;

    int b  = wid / (24 * 128);
    int r  = wid % (24 * 128);
    int o0 = (r >> 7) * 64;                           // 24 O-tiles of 64
    int n0 = (r & 127) * 32;                          // 128 N-tiles of 32

    const float* xb = x + (size_t)b * CDIM * NPIX;
    v8f acc[4][2] = {};
    for (int c0 = 0; c0 < CDIM; c0 += 4) {
        int kk = c0 + 2 * h;
        v2f bf0, bf1;
        bf0.x = xb[(size_t)kk * NPIX + n0 + m];
        bf0.y = xb[(size_t)(kk + 1) * NPIX + n0 + m];
        bf1.x = xb[(size_t)kk * NPIX + n0 + 16 + m];
        bf1.y = xb[(size_t)(kk + 1) * NPIX + n0 + 16 + m];
#pragma unroll
        for (int t = 0; t < 4; ++t) {
            const float* wrow = wg + (size_t)(o0 + t * 16 + m) * CDIM + kk;
            v2f af; af.x = wrow[0]; af.y = wrow[1];
            acc[t][0] = wmma_f32(af, bf0, acc[t][0]);
            acc[t][1] = wmma_f32(af, bf1, acc[t][1]);
        }
    }
    float sc0 = s1[b * NPIX + n0 + m];
    float sc1 = s1[b * NPIX + n0 + 16 + m];
#pragma unroll
    for (int t = 0; t < 4; ++t)
#pragma unroll
        for (int rr = 0; rr < 8; ++rr) {
            int orow = o0 + t * 16 + rr + 8 * h;
            size_t base = ((size_t)b * OQKV + orow) * NPIX + n0 + m;
            qkv[base]      = acc[t][0][rr] * sc0;
            qkv[base + 16] = acc[t][1][rr] * sc1;
        }
}

// ---------------------------------------------------------------- kernel 3
// per (b, head): softmax(k over n), context = k_sm @ v^T, softmax(q over d),
// out = ctx^T @ q * scale.   One 256-thread WG (8 waves) per (b,head).
__global__ __launch_bounds__(256) void attn(const float* __restrict__ qkv,
                                            float* __restrict__ attnout) {
    __shared__ float kmax[DHEAD];
    __shared__ float kinv[DHEAD];
    __shared__ float ctx[DHEAD * DHEAD];              // ctx[d][e]

    int b = blockIdx.x >> 3, head = blockIdx.x & 7;
    const float* qb = qkv + ((size_t)b * OQKV + head * DHEAD) * NPIX;
    const float* kb = qkv + ((size_t)b * OQKV + 512  + head * DHEAD) * NPIX;
    const float* vb = qkv + ((size_t)b * OQKV + 1024 + head * DHEAD) * NPIX;

    int wv = threadIdx.x >> 5, lane = threadIdx.x & 31;
    int m = lane & 15, h = lane >> 4;

    // ---- phase 1: k row stats (softmax over n=4096), 8 rows per wave
    for (int rr = 0; rr < 8; ++rr) {
        int d = wv * 8 + rr;
        const float* krow = kb + (size_t)d * NPIX;
        float mx = -INFINITY;
        for (int n = lane; n < NPIX; n += 32) mx = fmaxf(mx, krow[n]);
#pragma unroll
        for (int off = 16; off; off >>= 1) mx = fmaxf(mx, __shfl_xor(mx, off, 32));
        float sm = 0.f;
        for (int n = lane; n < NPIX; n += 32) sm += __expf(krow[n] - mx);
#pragma unroll
        for (int off = 16; off; off >>= 1) sm += __shfl_xor(sm, off, 32);
        if (lane == 0) { kmax[d] = mx; kinv[d] = 1.0f / sm; }
    }
    __syncthreads();

    // ---- phase 2: context[d][e] = sum_n ksm[d][n] * v[e][n]; K = 4096
    {
        int d0  = (wv >> 1) * 16;                     // 4 M(d)-tiles over 8 waves
        int e0a = (wv & 1) * 32, e0b = e0a + 16;      // 2 N(e)-tiles per wave
        int drow = d0 + m;
        float rmax = kmax[drow], rinv = kinv[drow];
        const float* krow  = kb + (size_t)drow * NPIX;
        const float* vrowA = vb + (size_t)(e0a + m) * NPIX;
        const float* vrowB = vb + (size_t)(e0b + m) * NPIX;
        v8f acc0 = {}, acc1 = {};
        for (int k0 = 0; k0 < NPIX; k0 += 4) {
            int kk = k0 + 2 * h;
            v2f af;
            af.x = __expf(krow[kk]     - rmax) * rinv;
            af.y = __expf(krow[kk + 1] - rmax) * rinv;
            v2f b0; b0.x = vrowA[kk]; b0.y = vrowA[kk + 1];
            v2f b1; b1.x = vrowB[kk]; b1.y = vrowB[kk + 1];
            acc0 = wmma_f32(af, b0, acc0);
            acc1 = wmma_f32(af, b1, acc1);
        }
#pragma unroll
        for (int rr = 0; rr < 8; ++rr) {
            int d = d0 + rr + 8 * h;
            ctx[d * DHEAD + e0a + m] = acc0[rr];
            ctx[d * DHEAD + e0b + m] = acc1[rr];
        }
    }
    __syncthreads();

    // ---- phase 3: out[e][n] = scale * sum_d ctx[d][e] * qsm[d][n]; K = 64
    const float scale = 0.125f;                        // 64^-0.5
    for (int nt = wv; nt < 256; nt += 8) {
        int n0  = nt * 16;
        int col = n0 + m;                              // lane pair (l, l+16) shares col
        int dbase = h * 32;                            // each half covers 32 d values
        float cmx = -INFINITY;
        for (int dd = 0; dd < 32; ++dd)
            cmx = fmaxf(cmx, qb[(size_t)(dbase + dd) * NPIX + col]);
        cmx = fmaxf(cmx, __shfl_xor(cmx, 16, 32));
        float cs = 0.f;
        for (int dd = 0; dd < 32; ++dd)
            cs += __expf(qb[(size_t)(dbase + dd) * NPIX + col] - cmx);
        cs += __shfl_xor(cs, 16, 32);
        float cinv = scale / cs;

        v8f acc[4] = {};
        for (int k0 = 0; k0 < DHEAD; k0 += 4) {
            int kk = k0 + 2 * h;
            v2f bf;
            bf.x = __expf(qb[(size_t)kk * NPIX + col] - cmx) * cinv;
            bf.y = __expf(qb[(size_t)(kk + 1) * NPIX + col] - cmx) * cinv;
#pragma unroll
            for (int t = 0; t < 4; ++t) {              // 4 M(e)-tiles share B frag
                v2f af;
                af.x = ctx[kk * DHEAD + t * 16 + m];
                af.y = ctx[(kk + 1) * DHEAD + t * 16 + m];
                acc[t] = wmma_f32(af, bf, acc[t]);
            }
        }
#pragma unroll
        for (int t = 0; t < 4; ++t)
#pragma unroll
            for (int rr = 0; rr < 8; ++rr) {
                int e = t * 16 + rr + 8 * h;
                attnout[((size_t)b * CDIM + head * DHEAD + e) * NPIX + n0 + m] = acc[t][rr];
            }
    }
}

// ---------------------------------------------------------------- kernel 4
// out[b][o][n] = sum_c w_out[o][c] * attn[b][c][n] + b_out[o]   (raw, pre-norm)
// wave tile: 64(O) x 32(N), K=512
__global__ __launch_bounds__(256) void gemm_out(const float* __restrict__ attnin,
                                                const float* __restrict__ wout,
                                                const float* __restrict__ bout,
                                                float* __restrict__ out) {
    int wid  = blockIdx.x * 8 + (threadIdx.x >> 5);   // 16384 waves
    int lane = threadIdx.x & 31;
    int m = lane & 15, h = lane >> 4;

    int b  = wid / (8 * 128);
    int r  = wid % (8 * 128);
    int o0 = (r >> 7) * 64;                           // 8 O-tiles of 64
    int n0 = (r & 127) * 32;                          // 128 N-tiles of 32

    const float* ab = attnin + (size_t)b * CDIM * NPIX;
    v8f acc[4][2] = {};
    for (int c0 = 0; c0 < CDIM; c0 += 4) {
        int kk = c0 + 2 * h;
        v2f bf0, bf1;
        bf0.x = ab[(size_t)kk * NPIX + n0 + m];
        bf0.y = ab[(size_t)(kk + 1) * NPIX + n0 + m];
        bf1.x = ab[(size_t)kk * NPIX + n0 + 16 + m];
        bf1.y = ab[(size_t)(kk + 1) * NPIX + n0 + 16 + m];
#pragma unroll
        for (int t = 0; t < 4; ++t) {
            const float* wrow = wout + (size_t)(o0 + t * 16 + m) * CDIM + kk;
            v2f af; af.x = wrow[0]; af.y = wrow[1];
            acc[t][0] = wmma_f32(af, bf0, acc[t][0]);
            acc[t][1] = wmma_f32(af, bf1, acc[t][1]);
        }
    }
#pragma unroll
    for (int t = 0; t < 4; ++t)
#pragma unroll
        for (int rr = 0; rr < 8; ++rr) {
            int orow = o0 + t * 16 + rr + 8 * h;
            float bb = bout[orow];
            size_t base = ((size_t)b * CDIM + orow) * NPIX + n0 + m;
            out[base]      = acc[t][0][rr] + bb;
            out[base + 16] = acc[t][1][rr] + bb;
        }
}

// ---------------------------------------------------------------- kernel 5
// in-place rmsnorm with gamma2
__global__ __launch_bounds__(256) void norm2(float* __restrict__ out,
                                             const float* __restrict__ gamma2) {
    int p = blockIdx.x * 256 + threadIdx.x;           // 65536 pixels
    int b = p >> 12, n = p & (NPIX - 1);
    float* ob = out + (size_t)b * CDIM * NPIX + n;
    float acc = 0.f;
    for (int c = 0; c < CDIM; ++c) {
        float v = ob[(size_t)c * NPIX];
        acc = fmaf(v, v, acc);
    }
    float s = SQRT_C / fmaxf(sqrtf(acc), EPS);
    for (int c = 0; c < CDIM; ++c)
        ob[(size_t)c * NPIX] *= s * gamma2[c];
}

extern "C" void kernel_launch(void* const* d_in, const int* in_sizes, int n_in,
                              void* d_out, int out_size, void* d_ws, size_t ws_size,
                              hipStream_t stream) {
    const float* x      = (const float*)d_in[0];
    const float* gamma1 = (const float*)d_in[1];
    const float* wqkv   = (const float*)d_in[2];
    const float* wout   = (const float*)d_in[3];
    const float* bout   = (const float*)d_in[4];
    const float* gamma2 = (const float*)d_in[5];
    float* out = (float*)d_out;

    float* ws  = (float*)d_ws;
    float* wg   = ws;                                  // 1536*512       = 786432
    float* s1   = ws + 786432;                         // 65536
    float* qkv  = ws + 851968;                         // 16*1536*4096   = 100663296
    float* attw = ws + 101515264;                      // 16*512*4096    = 33554432

    prep_wg <<<(OQKV * CDIM) / 256, 256, 0, stream>>>(wqkv, gamma1, wg);
    norm1   <<<(NB * NPIX) / 256,   256, 0, stream>>>(x, s1);
    gemm_qkv<<<(NB * 24 * 128) / 8, 256, 0, stream>>>(x, wg, s1, qkv);
    attn    <<<NB * HEADS,          256, 0, stream>>>(qkv, attw);
    gemm_out<<<(NB * 8 * 128) / 8,  256, 0, stream>>>(attw, wout, bout, out);
    norm2   <<<(NB * NPIX) / 256,   256, 0, stream>>>(out, gamma2);
}